// GaussianRenderer_14190571946096
// MI455X (gfx1250) — compile-verified
//
#include <hip/hip_runtime.h>
#include <math.h>

// CDNA5 / gfx1250 Gaussian-splat renderer.
//
// Key idea: the per-(pixel,gaussian) exponent P = -0.5*(ia*d0^2 + 2*ib*d0*d1 + ic*d1^2)
// + log(opacity) is bilinear:  P(p,g) = sum_k gfeat[g][k] * pixfeat[p][k], k<6 (pad to 8).
//   pixfeat = [x^2, x*y, y^2, x, y, 1, 0, 0]
//   gfeat   = [-0.5*ia, -ib, -0.5*ic, ia*mx+ib*my, ib*mx+ic*my, const, 0, 0]
//   const   = -0.5*(ia*mx^2 + 2*ib*mx*my + ic*my^2) + log(op)   (or -1e30 if invalid)
// Two V_WMMA_F32_16X16X4_F32 per 16-gaussian x 16-pixel tile evaluate P on the matrix
// pipe; VALU only does exp + sequential alpha blend. Orientation M=gaussians, N=pixels
// means each lane of the 16x16 f32 D tile holds 8 *consecutive sorted gaussians* of one
// pixel (low half: g0..7, high half: g8..15), so the transmittance cumprod is a per-lane
// register loop + one cross-half shuffle combine per chunk.
//
// All sorted gaussian data (2048 x 8 feature floats + 2048 x float4 colors = 96 KB)
// lives in LDS (320 KB/WGP -> 3 workgroups resident).

typedef __attribute__((ext_vector_type(2))) float v2f;
typedef __attribute__((ext_vector_type(8))) float v8f;

#define NG     2048
#define IMG_H  128
#define IMG_W  128

// ---------------------------------------------------------------- preprocess
__global__ void __launch_bounds__(256) k_preprocess(
    const float* __restrict__ means3D, const float* __restrict__ covs3d,
    const float* __restrict__ colors,  const float* __restrict__ opac,
    const float* __restrict__ Kc,      const float* __restrict__ Rm,
    const float* __restrict__ tv,
    float* __restrict__ depth, float* __restrict__ gfeat, float4* __restrict__ col4)
{
    int g = blockIdx.x * blockDim.x + threadIdx.x;
    if (g >= NG) return;

    float m0 = means3D[3*g+0], m1 = means3D[3*g+1], m2 = means3D[3*g+2];
    float R00=Rm[0],R01=Rm[1],R02=Rm[2];
    float R10=Rm[3],R11=Rm[4],R12=Rm[5];
    float R20=Rm[6],R21=Rm[7],R22=Rm[8];

    // cam = R*m + t   (reference: means3D @ R.T + t)
    float c0 = R00*m0 + R01*m1 + R02*m2 + tv[0];
    float c1 = R10*m0 + R11*m1 + R12*m2 + tv[1];
    float c2 = R20*m0 + R21*m1 + R22*m2 + tv[2];
    float dep = fmaxf(c2, 1.0f);

    // screen = K*cam ; means2D = screen.xy / screen.z
    float s0 = Kc[0]*c0 + Kc[1]*c1 + Kc[2]*c2;
    float s1 = Kc[3]*c0 + Kc[4]*c1 + Kc[5]*c2;
    float s2 = Kc[6]*c0 + Kc[7]*c1 + Kc[8]*c2;
    float mx = s0 / s2, my = s1 / s2;

    float fx = Kc[0], fy = Kc[4], cx = Kc[2], cy = Kc[5];
    float tz = c2, tz2 = tz*tz;
    float tanx = 0.5f*IMG_W/fx, tany = 0.5f*IMG_H/fy;
    float limxp = (IMG_W-cx)/fx + 0.3f*tanx, limxn = cx/fx + 0.3f*tanx;
    float limyp = (IMG_H-cy)/fy + 0.3f*tany, limyn = cy/fy + 0.3f*tany;
    float txc = tz * fminf(fmaxf(c0/tz, -limxn), limxp);
    float tyc = tz * fminf(fmaxf(c1/tz, -limyn), limyp);

    float J00 = fx/tz, J02 = -fx*txc/tz2;
    float J11 = fy/tz, J12 = -fy*tyc/tz2;

    // V = R * C * R^T
    const float* C = covs3d + 9*g;
    float RC[9];
    #pragma unroll
    for (int i = 0; i < 3; ++i) {
        float ri0 = Rm[3*i+0], ri1 = Rm[3*i+1], ri2 = Rm[3*i+2];
        RC[3*i+0] = ri0*C[0] + ri1*C[3] + ri2*C[6];
        RC[3*i+1] = ri0*C[1] + ri1*C[4] + ri2*C[7];
        RC[3*i+2] = ri0*C[2] + ri1*C[5] + ri2*C[8];
    }
    float V[9];
    #pragma unroll
    for (int i = 0; i < 3; ++i) {
        #pragma unroll
        for (int j = 0; j < 3; ++j)
            V[3*i+j] = RC[3*i+0]*Rm[3*j+0] + RC[3*i+1]*Rm[3*j+1] + RC[3*i+2]*Rm[3*j+2];
    }

    // cov2 = J V J^T  with J = [[J00,0,J02],[0,J11,J12]]
    float a0 = J00*V[0] + J02*V[6];          // row0 of J*V
    float a1 = J00*V[1] + J02*V[7];
    float a2 = J00*V[2] + J02*V[8];
    float b0 = J11*V[3] + J12*V[6];          // row1 of J*V
    float b1 = J11*V[4] + J12*V[7];
    float b2 = J11*V[5] + J12*V[8];
    float c00 = a0*J00 + a2*J02 + 1e-4f;
    float c01 = a1*J11 + a2*J12;
    float c10 = b0*J00 + b2*J02;
    float c11 = b1*J11 + b2*J12 + 1e-4f;

    float det = fmaxf(c00*c11 - c01*c10, 1e-4f);
    float ia  =  c11 / det;
    float ic  =  c00 / det;
    float ib  = -(c01 + c10) * 0.5f / det;

    bool valid = (dep > 1.0f) && (dep < 50.0f);
    float op   = opac[g];
    float kc   = -0.5f*(ia*mx*mx + 2.0f*ib*mx*my + ic*my*my)
               + (valid ? __logf(op) : -1e30f);

    float* gf = gfeat + 8*g;
    gf[0] = -0.5f*ia;       gf[1] = -ib;
    gf[2] = -0.5f*ic;       gf[3] = ia*mx + ib*my;
    gf[4] = ib*mx + ic*my;  gf[5] = kc;
    gf[6] = 0.0f;           gf[7] = 0.0f;

    depth[g] = dep;
    col4[g]  = make_float4(colors[3*g+0], colors[3*g+1], colors[3*g+2], 0.0f);
}

// ------------------------------------------------------- stable rank + scatter
// rank[g] = #{ j : depth[j] < depth[g]  or (depth[j]==depth[g] and j<g) }
// (matches jnp.argsort stable ordering); then scatter into sorted arrays.
__global__ void __launch_bounds__(256) k_rank_scatter(
    const float* __restrict__ depth, const float* __restrict__ gfeat,
    const float4* __restrict__ col4,
    float* __restrict__ sortedG, float4* __restrict__ sortedC)
{
    __shared__ float sd[256];
    int g = blockIdx.x * 256 + threadIdx.x;
    float dg = depth[g];
    int rank = 0;
    for (int tile = 0; tile < NG; tile += 256) {
        __syncthreads();
        sd[threadIdx.x] = depth[tile + threadIdx.x];
        __syncthreads();
        #pragma unroll 8
        for (int j = 0; j < 256; ++j) {
            float dj = sd[j];
            int   jj = tile + j;
            rank += (dj < dg) || (dj == dg && jj < g);
        }
    }
    #pragma unroll
    for (int k = 0; k < 8; ++k) sortedG[rank*8 + k] = gfeat[g*8 + k];
    sortedC[rank] = col4[g];
}

// ------------------------------------------------------------------- render
// 256 threads = 8 waves; wave w owns pixels [blockIdx*128 + w*16, +16).
// Per chunk of 16 sorted gaussians: 2x V_WMMA_F32_16X16X4_F32 (K padded 6->8),
// then per-lane 8-step inclusive-cumprod blend + cross-half shuffle combine.
__global__ void __launch_bounds__(256) k_render(
    const float* __restrict__ sortedG, const float4* __restrict__ sortedC,
    float* __restrict__ out)
{
    extern __shared__ float smem[];
    float* sG = smem;            // NG*8 floats (64 KB)
    float* sC = smem + NG*8;     // NG*4 floats (32 KB)

    int tid = threadIdx.x;
    {   // cooperative LDS fill (float4 granularity)
        const float4* srcG = (const float4*)sortedG;
        float4* dstG = (float4*)sG;
        for (int i = tid; i < NG*2; i += 256) dstG[i] = srcG[i];
        float4* dstC = (float4*)sC;
        for (int i = tid; i < NG;   i += 256) dstC[i] = sortedC[i];
    }
    __syncthreads();

    int wave = tid >> 5;          // wave32
    int lane = tid & 31;
    int half = lane >> 4;         // 0: K rows 0/1 (and gaussians 0..7 of D)
    int lid  = lane & 15;

    int pix = blockIdx.x * 128 + wave * 16 + lid;
    float px = (float)(pix & (IMG_W - 1));
    float py = (float)(pix >> 7);

    // B = pixel features, 4x16 (KxN), N = pixel column = lid.
    // Assumed layout (mirrors A/C tables): VGPR0 lanes0-15 = row K0, lanes16-31 = K2;
    // VGPR1 lanes0-15 = K1, lanes16-31 = K3.
    float f0 = px*px, f1 = px*py, f2 = py*py, f3 = px, f4 = py, f5 = 1.0f;
    v2f B1, B2;
    B1.x = half ? f2   : f0;   B1.y = half ? f3   : f1;
    B2.x = half ? 0.0f : f4;   B2.y = half ? 0.0f : f5;

    float Tpix = 1.0f;
    float accR = 0.0f, accG = 0.0f, accB = 0.0f;

    for (int chunk = 0; chunk < NG/16; ++chunk) {
        // A = gaussian features 16x4: lane L<16 -> row M=L gets K{0,1}; L>=16 -> K{2,3}.
        int n = chunk*16 + lid;
        const float* gp = sG + n*8 + half*2;
        v2f A1; A1.x = gp[0]; A1.y = gp[1];
        v2f A2; A2.x = gp[4]; A2.y = gp[5];   // high half reads padded zeros

        v8f Cz = {};
        v8f P = __builtin_amdgcn_wmma_f32_16x16x4_f32(
                    false, A1, false, B1, (short)0, Cz, false, false);
        P     = __builtin_amdgcn_wmma_f32_16x16x4_f32(
                    false, A2, false, B2, (short)0, P,  false, false);

        // D layout: lane holds pixel N=lid; VGPR i = gaussian M = half*8 + i.
        float Trel = 1.0f, aR = 0.0f, aG = 0.0f, aB = 0.0f;
        int gbase = chunk*16 + half*8;
        #pragma unroll
        for (int i = 0; i < 8; ++i) {
            float Pv    = fminf(P[i], 80.0f);
            float alpha = fminf(__expf(Pv), 0.999f);
            Trel *= (1.0f - alpha);
            float w = alpha * Trel;              // reference uses INCLUSIVE cumprod
            const float* cp = sC + (gbase + i)*4;
            aR += w * cp[0]; aG += w * cp[1]; aB += w * cp[2];
        }

        // Combine halves: relative partials -> absolute, identical on all lanes.
        int lo = lid, hi = lid + 16;
        float TrelL = __shfl(Trel, lo);
        float TrelH = __shfl(Trel, hi);
        float RL = __shfl(aR, lo), GL = __shfl(aG, lo), BL = __shfl(aB, lo);
        float RH = __shfl(aR, hi), GH = __shfl(aG, hi), BH = __shfl(aB, hi);
        accR += Tpix * (RL + TrelL * RH);
        accG += Tpix * (GL + TrelL * GH);
        accB += Tpix * (BL + TrelL * BH);
        Tpix *= TrelL * TrelH;

        // Exact-ish early out (wave-uniform -> EXEC stays all-1s for WMMA).
        if (__ballot(Tpix > 1e-30f) == 0) break;
    }

    if (lane < 16) {
        float* o = out + pix*3;
        o[0] = accR; o[1] = accG; o[2] = accB;
    }
}

// ---------------------------------------------------------------- launcher
extern "C" void kernel_launch(void* const* d_in, const int* in_sizes, int n_in,
                              void* d_out, int out_size, void* d_ws, size_t ws_size,
                              hipStream_t stream) {
    const float* means3D = (const float*)d_in[0];
    const float* covs3d  = (const float*)d_in[1];
    const float* colors  = (const float*)d_in[2];
    const float* opac    = (const float*)d_in[3];
    const float* Kc      = (const float*)d_in[4];
    const float* Rm      = (const float*)d_in[5];
    const float* tv      = (const float*)d_in[6];
    // d_in[7] = pixels: exact meshgrid(x,y), regenerated analytically in k_render.

    char* ws = (char*)d_ws;
    float*  depth   = (float*) (ws);                                   //  8 KB
    float*  gfeat   = (float*) (ws + 8192);                            // 64 KB
    float4* col4    = (float4*)(ws + 8192 + 65536);                    // 32 KB
    float*  sortedG = (float*) (ws + 8192 + 65536 + 32768);            // 64 KB
    float4* sortedC = (float4*)(ws + 8192 + 65536 + 32768 + 65536);    // 32 KB

    k_preprocess<<<NG/256, 256, 0, stream>>>(means3D, covs3d, colors, opac,
                                             Kc, Rm, tv, depth, gfeat, col4);
    k_rank_scatter<<<NG/256, 256, 0, stream>>>(depth, gfeat, col4, sortedG, sortedC);

    size_t lds = (size_t)(NG*8 + NG*4) * sizeof(float);   // 96 KB / workgroup
    k_render<<<(IMG_H*IMG_W)/128, 256, lds, stream>>>(sortedG, sortedC, (float*)d_out);
}